// Model_86431921865161
// MI455X (gfx1250) — compile-verified
//
#include <hip/hip_runtime.h>
#include <math.h>

#define BATCH 256
#define NFEAT 640
#define GIN   128
#define GHID  128
#define NEXP  8
#define MDIM  512   // M_IN == M_HID == M_OUT == 512

typedef __attribute__((ext_vector_type(2))) float v2f;
typedef __attribute__((ext_vector_type(4))) float v4f;
typedef __attribute__((ext_vector_type(8))) float v8f;

__device__ __forceinline__ float elu1(float x) { return x > 0.f ? x : expm1f(x); }

// ---------------------------------------------------------------- normalize
__global__ void k_norm(const float* __restrict__ x, const float* __restrict__ Xn,
                       float* __restrict__ xn, int n) {
    int i = blockIdx.x * blockDim.x + threadIdx.x;
    if (i < n) {
        int f = i % NFEAT;
        xn[i] = (x[i] - Xn[f]) / Xn[NFEAT + f];
    }
}

// ---------------------------------------------------------------- gating MLP
// one block (128 threads) per batch column
__global__ void k_gate(const float* __restrict__ xn,
                       const float* __restrict__ GW1, const float* __restrict__ Gb1,
                       const float* __restrict__ GW2, const float* __restrict__ Gb2,
                       const float* __restrict__ GW3, const float* __restrict__ Gb3,
                       float* __restrict__ gate) {
    __shared__ float s0[GHID], s1[GHID], logit[NEXP];
    int b = blockIdx.x, t = threadIdx.x;
    s0[t] = xn[b * NFEAT + t];           // first G_IN=128 features
    __syncthreads();
    float acc = Gb1[t];
    for (int i = 0; i < GIN; ++i) acc += GW1[t * GIN + i] * s0[i];
    s1[t] = elu1(acc);
    __syncthreads();
    acc = Gb2[t];
    for (int i = 0; i < GHID; ++i) acc += GW2[t * GHID + i] * s1[i];
    __syncthreads();
    s0[t] = elu1(acc);
    __syncthreads();
    if (t < NEXP) {
        float l = Gb3[t];
        for (int i = 0; i < GHID; ++i) l += GW3[t * GHID + i] * s0[i];
        logit[t] = l;
    }
    __syncthreads();
    if (t == 0) {
        float mx = logit[0];
        for (int e = 1; e < NEXP; ++e) mx = fmaxf(mx, logit[e]);
        float s = 0.f;
        float ex[NEXP];
        for (int e = 0; e < NEXP; ++e) { ex[e] = __expf(logit[e] - mx); s += ex[e]; }
        float inv = 1.f / s;
        for (int e = 0; e < NEXP; ++e) gate[e * BATCH + b] = ex[e] * inv;
    }
}

// ---------------------------------------------------------------- weight mixing (HBM-write bound)
// w[b, :] = sum_e gate[e,b] * EW[e, :]
// EW (24 MB total) stays L2-resident; the 805 MB output stream uses
// non-temporal stores so it doesn't evict the weights from L2.
__global__ __launch_bounds__(256) void k_mixw(const float* __restrict__ gate,
                                              const float* __restrict__ EW,
                                              float* __restrict__ out) {
    int b = blockIdx.y;
    size_t idx = ((size_t)blockIdx.x * blockDim.x + threadIdx.x) * 4;  // < 512*512
    float g[NEXP];
#pragma unroll
    for (int e = 0; e < NEXP; ++e) g[e] = gate[e * BATCH + b];
    v4f acc = {0.f, 0.f, 0.f, 0.f};
#pragma unroll
    for (int e = 0; e < NEXP; ++e) {
        v4f w = *(const v4f*)(EW + (size_t)e * MDIM * MDIM + idx);
        acc += g[e] * w;
    }
    __builtin_nontemporal_store(acc, (v4f*)(out + (size_t)b * MDIM * MDIM + idx));
}

// ---------------------------------------------------------------- bias mixing (tiny)
__global__ void k_mixb(const float* __restrict__ gate, const float* __restrict__ Eb,
                       float* __restrict__ out) {
    int i = blockIdx.x * blockDim.x + threadIdx.x;  // b*512 + h
    int b = i >> 9, h = i & 511;
    float acc = 0.f;
#pragma unroll
    for (int e = 0; e < NEXP; ++e) acc += gate[e * BATCH + b] * Eb[e * MDIM + h];
    out[i] = acc;
}

// ---------------------------------------------------------------- expert layer: f32 WMMA
// OUT[b, h] = act( sum_e gate[e,b] * (EW[e] @ in_b)[h] + bmix[b,h] )
// One block per 16x16 (h x batch) output tile; wave e computes expert e's
// contribution (dual accumulators to break the WMMA RAW chain), scales it by
// gate[e,b], then the block reduces across experts through LDS.
__global__ __launch_bounds__(256) void k_layer(const float* __restrict__ EW,
                                               const float* __restrict__ IN,
                                               const float* __restrict__ gate,
                                               const float* __restrict__ bmix,
                                               const float* __restrict__ Yn,
                                               float* __restrict__ OUT,
                                               int stride, int inOff, int mode) {
    const int K = MDIM;
    __shared__ float red[NEXP][256];

    int e    = threadIdx.x >> 5;   // wave index == expert index
    int lane = threadIdx.x & 31;
    int tile = blockIdx.x;
    int tm = tile & 31;            // 32 h-tiles
    int tn = tile >> 5;            // 16 batch-tiles
    int l15 = lane & 15;
    int hi  = lane >> 4;

    int rowA = tm * 16 + l15;      // output-feature row for A operand
    int bcol = tn * 16 + l15;      // batch column for B operand / C layout

    const float* ap = EW + (size_t)e * MDIM * K + (size_t)rowA * K + 2 * hi;
    const float* bp = IN + (size_t)bcol * stride + inOff + 2 * hi;

    v8f c0 = {0.f, 0.f, 0.f, 0.f, 0.f, 0.f, 0.f, 0.f};
    v8f c1 = {0.f, 0.f, 0.f, 0.f, 0.f, 0.f, 0.f, 0.f};
#pragma unroll 4
    for (int k = 0; k < K; k += 8) {
        v2f a0 = *(const v2f*)(ap + k);
        v2f b0 = *(const v2f*)(bp + k);
        v2f a1 = *(const v2f*)(ap + k + 4);
        v2f b1 = *(const v2f*)(bp + k + 4);
        c0 = __builtin_amdgcn_wmma_f32_16x16x4_f32(false, a0, false, b0,
                                                   (short)0, c0, false, false);
        c1 = __builtin_amdgcn_wmma_f32_16x16x4_f32(false, a1, false, b1,
                                                   (short)0, c1, false, false);
    }
    float gs = gate[e * BATCH + bcol];
#pragma unroll
    for (int r = 0; r < 8; ++r) red[e][r * 32 + lane] = gs * (c0[r] + c1[r]);
    __syncthreads();

    // cross-expert reduction + epilogue: thread t handles flat element t
    int t = threadIdx.x;           // t = r*32 + lane
    int rlane = t & 31;
    int rr    = t >> 5;
    int rhi   = rlane >> 4;
    int rl15  = rlane & 15;
    int o = tm * 16 + rr + 8 * rhi;     // output feature (C/D layout row)
    int b = tn * 16 + rl15;             // batch
    float v = bmix[(size_t)b * MDIM + o];
#pragma unroll
    for (int ee = 0; ee < NEXP; ++ee) v += red[ee][t];
    if (mode == 0) v = elu1(v);
    else           v = v * Yn[MDIM + o] + Yn[o];
    OUT[(size_t)b * MDIM + o] = v;
}

// ---------------------------------------------------------------- launch
extern "C" void kernel_launch(void* const* d_in, const int* in_sizes, int n_in,
                              void* d_out, int out_size, void* d_ws, size_t ws_size,
                              hipStream_t stream) {
    const float* x    = (const float*)d_in[0];
    const float* GW1  = (const float*)d_in[1];
    const float* Gb1  = (const float*)d_in[2];
    const float* GW2  = (const float*)d_in[3];
    const float* Gb2  = (const float*)d_in[4];
    const float* GW3  = (const float*)d_in[5];
    const float* Gb3  = (const float*)d_in[6];
    const float* EW1  = (const float*)d_in[7];
    const float* Eb1  = (const float*)d_in[8];
    const float* EW2  = (const float*)d_in[9];
    const float* Eb2  = (const float*)d_in[10];
    const float* EW3  = (const float*)d_in[11];
    const float* Eb3  = (const float*)d_in[12];
    const float* Xnorm = (const float*)d_in[13];
    const float* Ynorm = (const float*)d_in[14];

    float* out = (float*)d_out;
    float* y    = out;                       // 256*512
    float* gate = out + 131072;              // 8*256
    float* w0   = out + 133120;              // 256*512*512 each
    float* w1   = w0 + (size_t)67108864;
    float* w2   = w1 + (size_t)67108864;

    float* ws = (float*)d_ws;
    float* xn = ws;                          // 256*640 = 163840
    float* h1 = xn + 163840;                 // 256*512
    float* h2 = h1 + 131072;
    float* bm0 = h2 + 131072;                // 3 * 256*512
    float* bm1 = bm0 + 131072;
    float* bm2 = bm1 + 131072;

    k_norm<<<(BATCH * NFEAT + 255) / 256, 256, 0, stream>>>(x, Xnorm, xn, BATCH * NFEAT);
    k_gate<<<BATCH, GHID, 0, stream>>>(xn, GW1, Gb1, GW2, Gb2, GW3, Gb3, gate);

    dim3 gm(256, BATCH);   // 256 float4-chunks of 262144 floats  x  256 batches
    k_mixw<<<gm, 256, 0, stream>>>(gate, EW1, w0);
    k_mixw<<<gm, 256, 0, stream>>>(gate, EW2, w1);
    k_mixw<<<gm, 256, 0, stream>>>(gate, EW3, w2);

    k_mixb<<<512, 256, 0, stream>>>(gate, Eb1, bm0);
    k_mixb<<<512, 256, 0, stream>>>(gate, Eb2, bm1);
    k_mixb<<<512, 256, 0, stream>>>(gate, Eb3, bm2);

    // expert layers: one block per 16x16 tile, 8 expert-waves per block
    k_layer<<<512, 256, 0, stream>>>(EW1, xn, gate, bm0, Ynorm, h1, NFEAT, GIN, 0);
    k_layer<<<512, 256, 0, stream>>>(EW2, h1, gate, bm1, Ynorm, h2, MDIM, 0, 0);
    k_layer<<<512, 256, 0, stream>>>(EW3, h2, gate, bm2, Ynorm, y,  MDIM, 0, 1);

    (void)in_sizes; (void)n_in; (void)out_size; (void)ws_size;
}